// BEVDeformableTokenizer_v2_11433202942361
// MI455X (gfx1250) — compile-verified
//
#include <hip/hip_runtime.h>
#include <hip/hip_bf16.h>
#include <cstdint>
#include <cstddef>

// ---------------------------------------------------------------------------
// BEV deformable transformer, 4 layers, fp32, MI455X (gfx1250, wave32).
// All GEMMs + both attention matmuls use V_WMMA_F32_16X16X4_F32 (fp32 WMMA).
// ---------------------------------------------------------------------------

typedef __attribute__((ext_vector_type(2))) float v2f;
typedef __attribute__((ext_vector_type(8))) float v8f;

#define B_     2
#define NCAM_  6
#define BN_    12
#define NT_    2500
#define ML_    1500
#define D_     256
#define HH_    8
#define HD_    32
#define LVL_   4
#define PPT_   8      // points per level
#define SS_    4      // ref samples
#define DFF_   1024
#define LIN_   3740
#define MROWS  (B_ * NT_)     // 5000
#define RBROWS (BN_ * ML_)    // 18000
#define CTILES 157            // ceil(2500/16)

static __device__ __forceinline__ v8f wmma4(v2f a, v2f b, v8f c) {
  // D(16x16,f32) = A(16x4,f32) x B(4x16,f32) + C
  return __builtin_amdgcn_wmma_f32_16x16x4_f32(false, a, false, b, (short)0, c,
                                               false, false);
}

// ---------------------------------------------------------------------------
// Generic GEMM:  C[M,N] = A[M,K] * W[N,K]^T + bias   (optionally ReLU)
// One 16x16 output tile per wave; K stepped by 4 (one f32 WMMA per step).
// A-frag: lane L -> A[tm*16 + L%16][k + 2*(L/16) + {0,1}]  (contig float2)
// B-frag: lane L -> W[tn*16 + L%16][k + 2*(L/16) + {0,1}]  (contig float2)
// C-frag: lane L, reg v -> C[tm*16 + v + 8*(L/16)][tn*16 + L%16]
// ---------------------------------------------------------------------------
template <int ACT>
__global__ __launch_bounds__(256) void gemm_bias(
    const float* __restrict__ A, const float* __restrict__ W,
    const float* __restrict__ bias, float* __restrict__ C,
    int M, int N, int K, int ldc) {
  const int tilesN = (N + 15) >> 4;
  const int tilesM = (M + 15) >> 4;
  const int tile = blockIdx.x * 8 + (threadIdx.x >> 5);
  if (tile >= tilesM * tilesN) return;
  const int tm = tile / tilesN, tn = tile - tm * tilesN;
  const int lane = threadIdx.x & 31;
  const int hf = lane >> 4, lm = lane & 15;

  int ar = tm * 16 + lm; if (ar > M - 1) ar = M - 1;   // clamp; store guarded
  int wr = tn * 16 + lm; if (wr > N - 1) wr = N - 1;
  const float* Ap = A + (size_t)ar * K + 2 * hf;
  const float* Wp = W + (size_t)wr * K + 2 * hf;

  v8f c = {};
#pragma unroll 4
  for (int k = 0; k < K; k += 4) {
    v2f a = *(const v2f*)(Ap + k);
    v2f b = *(const v2f*)(Wp + k);
    c = wmma4(a, b, c);
  }

  const int coln = tn * 16 + lm;
  if (coln < N) {
    const float bv = bias[coln];
#pragma unroll
    for (int v = 0; v < 8; ++v) {
      const int rm = tm * 16 + v + 8 * hf;
      if (rm < M) {
        float val = c[v] + bv;
        if (ACT) val = fmaxf(val, 0.f);
        C[(size_t)rm * ldc + coln] = val;
      }
    }
  }
}

// ---------------------------------------------------------------------------
// x = tgt + query_pos (broadcast over batch)
// ---------------------------------------------------------------------------
__global__ __launch_bounds__(256) void add_pos(const float* __restrict__ t,
                                               const float* __restrict__ qp,
                                               float* __restrict__ x) {
  const int r = blockIdx.x;          // 0..4999
  const int m = (r < NT_) ? r : r - NT_;
  x[(size_t)r * D_ + threadIdx.x] =
      t[(size_t)r * D_ + threadIdx.x] + qp[(size_t)m * D_ + threadIdx.x];
}

// ---------------------------------------------------------------------------
// MHA: flash-style, per (b,h) x 16-query-row block. 8 waves split key tiles.
// qkv layout: [B*NT, 768], cols [q|k|v], channel = h*32 + d.
// ---------------------------------------------------------------------------
__global__ __launch_bounds__(256) void mha_attn(const float* __restrict__ qkv,
                                                float* __restrict__ o) {
  const int bh = blockIdx.y;                 // 0..15
  const int b = bh >> 3, h = bh & 7;
  const int r0 = blockIdx.x << 4;            // query row block
  const int wv = threadIdx.x >> 5;
  const int lane = threadIdx.x & 31;
  const int hf = lane >> 4, lm = lane & 15;

  __shared__ float s_m[8][16], s_s[8][16];
  __shared__ float s_M[16], s_S[16];
  __shared__ __align__(16) float s_p[8][16][18];   // per-wave P layout bounce
  __shared__ float s_o[8][16][33];                 // per-wave partial O

  const float scale = 0.17677669529663687f;        // 1/sqrt(32)
  const float* qbase = qkv + ((size_t)b * NT_) * 768 + h * 32;
  const float* kbase = qbase + 256;
  const float* vbase = qbase + 512;

  // Q fragments for this row block (shared by all tiles of this wave)
  int qrow = r0 + lm; if (qrow >= NT_) qrow = NT_ - 1;
  const float* qp = qbase + (size_t)qrow * 768 + 2 * hf;
  v2f qa[8];
#pragma unroll
  for (int kk = 0; kk < 8; ++kk) qa[kk] = *(const v2f*)(qp + kk * 4);

  // ---- pass 1: per-lane online row max / sumexp over this wave's tiles ----
  float m8[8], s8[8];
#pragma unroll
  for (int v = 0; v < 8; ++v) { m8[v] = -3.0e38f; s8[v] = 0.f; }

  for (int ct = wv; ct < CTILES; ct += 8) {
    const int col = (ct << 4) + lm;
    const int cc = col < NT_ ? col : NT_ - 1;
    const float* kp = kbase + (size_t)cc * 768 + 2 * hf;
    v8f c = {};
#pragma unroll
    for (int kk = 0; kk < 8; ++kk) {
      v2f kb = *(const v2f*)(kp + kk * 4);
      c = wmma4(qa[kk], kb, c);
    }
    if (col < NT_) {
#pragma unroll
      for (int v = 0; v < 8; ++v) {
        const float x = c[v] * scale;
        const float mn = fmaxf(m8[v], x);
        s8[v] = s8[v] * __expf(m8[v] - mn) + __expf(x - mn);
        m8[v] = mn;
      }
    }
  }
  // butterfly-merge (m,s) across the 16 lanes holding each row
#pragma unroll
  for (int off = 1; off < 16; off <<= 1) {
#pragma unroll
    for (int v = 0; v < 8; ++v) {
      const float mo = __shfl_xor(m8[v], off);
      const float so = __shfl_xor(s8[v], off);
      const float mn = fmaxf(m8[v], mo);
      s8[v] = s8[v] * __expf(m8[v] - mn) + so * __expf(mo - mn);
      m8[v] = mn;
    }
  }
  if (lm == 0) {
#pragma unroll
    for (int v = 0; v < 8; ++v) {
      s_m[wv][hf * 8 + v] = m8[v];
      s_s[wv][hf * 8 + v] = s8[v];
    }
  }
  __syncthreads();
  if (threadIdx.x < 16) {
    float M = -3.0e38f, S = 0.f;
    for (int w = 0; w < 8; ++w) {
      const float mo = s_m[w][threadIdx.x], so = s_s[w][threadIdx.x];
      const float mn = fmaxf(M, mo);
      S = S * __expf(M - mn) + so * __expf(mo - mn);
      M = mn;
    }
    s_M[threadIdx.x] = M;
    s_S[threadIdx.x] = S;
  }
  __syncthreads();

  // ---- pass 2: recompute S, normalize to P, O += P * V via WMMA ----------
  v8f o0 = {}, o1 = {};
  for (int ct = wv; ct < CTILES; ct += 8) {
    const int col = (ct << 4) + lm;
    const int cc = col < NT_ ? col : NT_ - 1;
    const float* kp = kbase + (size_t)cc * 768 + 2 * hf;
    v8f c = {};
#pragma unroll
    for (int kk = 0; kk < 8; ++kk) {
      v2f kb = *(const v2f*)(kp + kk * 4);
      c = wmma4(qa[kk], kb, c);
    }
#pragma unroll
    for (int v = 0; v < 8; ++v) {
      const int row = (hf << 3) + v;
      float p = 0.f;
      if (col < NT_) p = __expf(c[v] * scale - s_M[row]) / s_S[row];
      s_p[wv][row][lm] = p;   // C-layout -> LDS
    }
    asm volatile("s_wait_dscnt 0x0" ::: "memory");  // in-wave store->load
    const int tbase = ct << 4;
#pragma unroll
    for (int kk = 0; kk < 4; ++kk) {
      v2f pa = *(const v2f*)(&s_p[wv][lm][(kk << 2) + (hf << 1)]);  // A-layout
      const int t0 = tbase + (kk << 2) + (hf << 1);
      const int tc0 = t0 < NT_ ? t0 : NT_ - 1;
      const int tc1 = (t0 + 1) < NT_ ? (t0 + 1) : NT_ - 1;
      const float* vp0 = vbase + (size_t)tc0 * 768;
      const float* vp1 = vbase + (size_t)tc1 * 768;
      v2f vb0 = {vp0[lm], vp1[lm]};
      v2f vb1 = {vp0[16 + lm], vp1[16 + lm]};
      o0 = wmma4(pa, vb0, o0);
      o1 = wmma4(pa, vb1, o1);
    }
  }
#pragma unroll
  for (int v = 0; v < 8; ++v) {
    s_o[wv][(hf << 3) + v][lm] = o0[v];
    s_o[wv][(hf << 3) + v][16 + lm] = o1[v];
  }
  __syncthreads();
  for (int e = threadIdx.x; e < 512; e += 256) {
    const int row = e >> 5, cl = e & 31;
    float acc = 0.f;
#pragma unroll
    for (int w = 0; w < 8; ++w) acc += s_o[w][row][cl];
    const int m = r0 + row;
    if (m < NT_) o[((size_t)(b * NT_ + m)) * D_ + h * 32 + cl] = acc;
  }
}

// ---------------------------------------------------------------------------
// y = LayerNorm(a + t2) * g + bb     (one wave per 256-wide row)
// ---------------------------------------------------------------------------
__global__ __launch_bounds__(256) void residual_ln(
    const float* __restrict__ a, const float* __restrict__ t2,
    const float* __restrict__ g, const float* __restrict__ bb,
    float* __restrict__ y, int rows) {
  const int w = threadIdx.x >> 5, lane = threadIdx.x & 31;
  const int row = blockIdx.x * 8 + w;
  if (row >= rows) return;
  const float* pa = a + (size_t)row * D_;
  const float* pb = t2 + (size_t)row * D_;
  float v[8], s = 0.f, s2 = 0.f;
#pragma unroll
  for (int i = 0; i < 8; ++i) {
    const float t = pa[lane * 8 + i] + pb[lane * 8 + i];
    v[i] = t; s += t; s2 += t * t;
  }
#pragma unroll
  for (int off = 16; off; off >>= 1) {
    s += __shfl_xor(s, off);
    s2 += __shfl_xor(s2, off);
  }
  const float mean = s * (1.f / 256.f);
  const float var = s2 * (1.f / 256.f) - mean * mean;
  const float rstd = rsqrtf(var + 1e-5f);
  float* py = y + (size_t)row * D_;
#pragma unroll
  for (int i = 0; i < 8; ++i) {
    const int c = lane * 8 + i;
    py[c] = (v[i] - mean) * rstd * g[c] + bb[c];
  }
}

// ---------------------------------------------------------------------------
// per-camera gather: tgt_rb / q_rb / ref_rb
// ---------------------------------------------------------------------------
__global__ __launch_bounds__(256) void gather_rb(
    const float* __restrict__ tgt2, const float* __restrict__ qpos,
    const float* __restrict__ refp, const int* __restrict__ cidx,
    float* __restrict__ tgt_rb, float* __restrict__ q_rb,
    float* __restrict__ ref_rb) {
  const int r = blockIdx.x;                 // 0..17999
  const int cam = r / ML_, j = r - cam * ML_;
  const int tok = cidx[cam * ML_ + j];
  const int b = cam / NCAM_;
  const int t = threadIdx.x;
  const float tv = tgt2[((size_t)(b * NT_ + tok)) * D_ + t];
  tgt_rb[(size_t)r * D_ + t] = tv;
  q_rb[(size_t)r * D_ + t] = tv + qpos[(size_t)tok * D_ + t];
  if (t < SS_ * 2) ref_rb[r * 8 + t] = refp[((size_t)cam * NT_ + tok) * 8 + t];
}

// ---------------------------------------------------------------------------
// Deformable sampling: one wave per (cam, token, head); lane = HD channel.
// 32 weights softmaxed via shuffles; 32 bilinear samples; corner fetches are
// fully coalesced 128B transactions (channel-contiguous val layout).
// ---------------------------------------------------------------------------
__global__ __launch_bounds__(256) void deform_sample(
    const float* __restrict__ val, const float* __restrict__ offb,
    const float* __restrict__ wtsb, const float* __restrict__ ref_rb,
    float* __restrict__ out) {
  const int r = blockIdx.x;                 // (cam, token)
  const int cam = r / ML_;
  const int h = threadIdx.x >> 5;
  const int lane = threadIdx.x & 31;

  // softmax over L*P = 32 weights (lane i holds weight i)
  float x = wtsb[(size_t)r * 256 + h * 32 + lane];
  float mx = x;
#pragma unroll
  for (int off = 16; off; off >>= 1) mx = fmaxf(mx, __shfl_xor(mx, off));
  const float e = __expf(x - mx);
  float sm = e;
#pragma unroll
  for (int off = 16; off; off >>= 1) sm += __shfl_xor(sm, off);
  const float wgt = e / sm;

  // lane i owns point i: l = i/8, p = i%8, s = p/2, pp = p%2
  const int l = lane >> 3, p = lane & 7, sp = p >> 1, pq = p & 1;
  const float* op = offb + (size_t)r * 512 + h * 128 + l * 16 + sp * 4 + pq * 2;
  const float invW = 1.0f / (float)(88 >> l);
  const float invH = 1.0f / (float)(32 >> l);
  const float lx = ref_rb[r * 8 + sp * 2 + 0] + op[0] * invW;
  const float ly = ref_rb[r * 8 + sp * 2 + 1] + op[1] * invH;

  const float* vb = val + (size_t)cam * LIN_ * 256 + h * 32 + lane;
  const int cstart[4] = {0, 2816, 3520, 3696};
  float acc = 0.f;
#pragma unroll
  for (int i = 0; i < 32; ++i) {
    const float w = __shfl(wgt, i);
    const float px = __shfl(lx, i);
    const float py = __shfl(ly, i);
    const int li = i >> 3;
    const int Hl = 32 >> li, Wl = 88 >> li, st = cstart[li];
    const float fx = px * (float)Wl - 0.5f;
    const float fy = py * (float)Hl - 0.5f;
    const float x0f = floorf(fx), y0f = floorf(fy);
    const float wx = fx - x0f, wy = fy - y0f;
    const int x0 = (int)x0f, y0 = (int)y0f;
    float v00 = 0.f, v01 = 0.f, v10 = 0.f, v11 = 0.f;
    const bool xa = (unsigned)x0 < (unsigned)Wl;
    const bool xb2 = (unsigned)(x0 + 1) < (unsigned)Wl;
    const bool ya = (unsigned)y0 < (unsigned)Hl;
    const bool yb = (unsigned)(y0 + 1) < (unsigned)Hl;
    if (xa && ya) v00 = vb[(size_t)(st + y0 * Wl + x0) * 256];
    if (xb2 && ya) v01 = vb[(size_t)(st + y0 * Wl + x0 + 1) * 256];
    if (xa && yb) v10 = vb[(size_t)(st + (y0 + 1) * Wl + x0) * 256];
    if (xb2 && yb) v11 = vb[(size_t)(st + (y0 + 1) * Wl + x0 + 1) * 256];
    acc += w * (v00 * (1.f - wx) * (1.f - wy) + v01 * wx * (1.f - wy) +
                v10 * (1.f - wx) * wy + v11 * wx * wy);
  }
  out[(size_t)r * 256 + h * 32 + lane] = acc;
}

// ---------------------------------------------------------------------------
// scatter-add rebatched rows into [B, NT, D] + counts; then normalize
// ---------------------------------------------------------------------------
__global__ __launch_bounds__(256) void scatter_add(
    const float* __restrict__ rb2, const int* __restrict__ cidx,
    float* __restrict__ acc, float* __restrict__ cnt) {
  const int r = blockIdx.x;
  const int cam = r / ML_, j = r - cam * ML_;
  const int tok = cidx[cam * ML_ + j];
  const int b = cam / NCAM_;
  atomicAdd(&acc[((size_t)(b * NT_ + tok)) * D_ + threadIdx.x],
            rb2[(size_t)r * D_ + threadIdx.x]);
  if (threadIdx.x == 0) atomicAdd(&cnt[b * NT_ + tok], 1.0f);
}

__global__ __launch_bounds__(256) void normalize_avg(
    const float* __restrict__ acc, const float* __restrict__ cnt,
    float* __restrict__ fin) {
  const int r = blockIdx.x;    // 0..4999
  const float inv = 1.0f / (cnt[r] + 0.001f);
  fin[(size_t)r * D_ + threadIdx.x] = acc[(size_t)r * D_ + threadIdx.x] * inv;
}

// ---------------------------------------------------------------------------
// host side
// ---------------------------------------------------------------------------
static inline void launch_gemm(const float* A, const float* W, const float* b,
                               float* C, int M, int N, int K, int ldc, int act,
                               hipStream_t s) {
  const int tiles = ((M + 15) / 16) * ((N + 15) / 16);
  const int blocks = (tiles + 7) / 8;
  if (act)
    gemm_bias<1><<<blocks, 256, 0, s>>>(A, W, b, C, M, N, K, ldc);
  else
    gemm_bias<0><<<blocks, 256, 0, s>>>(A, W, b, C, M, N, K, ldc);
}

extern "C" void kernel_launch(void* const* d_in, const int* in_sizes, int n_in,
                              void* d_out, int out_size, void* d_ws,
                              size_t ws_size, hipStream_t stream) {
  (void)in_sizes; (void)n_in; (void)out_size; (void)ws_size;

  const float* tgt_in = (const float*)d_in[0];
  const float* qpos   = (const float*)d_in[1];
  const float* refp   = (const float*)d_in[2];
  const float* src    = (const float*)d_in[3];
  const int*   cidx   = (const int*)d_in[4];
  const float* P_inw  = (const float*)d_in[5];   // [4,768,256]
  const float* P_inb  = (const float*)d_in[6];   // [4,768]
  const float* P_aow  = (const float*)d_in[7];   // [4,256,256]
  const float* P_aob  = (const float*)d_in[8];
  const float* P_n2g  = (const float*)d_in[9];
  const float* P_n2b  = (const float*)d_in[10];
  const float* P_Wv   = (const float*)d_in[11];
  const float* P_bv   = (const float*)d_in[12];
  const float* P_Woff = (const float*)d_in[13];  // [4,512,256]
  const float* P_boff = (const float*)d_in[14];
  const float* P_Ww   = (const float*)d_in[15];  // [4,256,256]
  const float* P_bw   = (const float*)d_in[16];
  const float* P_Wout = (const float*)d_in[17];
  const float* P_bout = (const float*)d_in[18];
  const float* P_n1g  = (const float*)d_in[19];
  const float* P_n1b  = (const float*)d_in[20];
  const float* P_W1   = (const float*)d_in[21];  // [4,1024,256]
  const float* P_b1   = (const float*)d_in[22];
  const float* P_W2   = (const float*)d_in[23];  // [4,256,1024]
  const float* P_b2   = (const float*)d_in[24];
  const float* P_n3g  = (const float*)d_in[25];
  const float* P_n3b  = (const float*)d_in[26];

  // workspace layout (floats) — ~208 MB total, stage-safe reuse noted
  constexpr size_t SZ_X   = (size_t)MROWS * D_;       // 1,280,000
  constexpr size_t SZ_QKV = (size_t)MROWS * 3 * D_;   // 3,840,000
  constexpr size_t SZ_RB  = (size_t)RBROWS * D_;      // 4,608,000
  constexpr size_t SZ_REF = (size_t)RBROWS * 8;
  constexpr size_t SZ_VAL = (size_t)BN_ * LIN_ * D_;  // 11,489,280 (reused: ffn hidden)
  constexpr size_t SZ_OFF = (size_t)RBROWS * 512;     //  9,216,000 (reused: t2_rb)

  float* w = (float*)d_ws;
  float* xb    = w; w += SZ_X;      // x = tgt+pos; later reused as tgt2
  float* qkvb  = w; w += SZ_QKV;
  float* attno = w; w += SZ_X;
  float* t2s   = w; w += SZ_X;
  float* tgtA  = w; w += SZ_X;
  float* tgtB  = w; w += SZ_X;
  float* tgtrb = w; w += SZ_RB;
  float* qrb   = w; w += SZ_RB;     // later reused as rb2 (LN output)
  float* refrb = w; w += SZ_REF;
  float* valb  = w; w += SZ_VAL;    // later reused as ffn hidden [5000,1024]
  float* offb  = w; w += SZ_OFF;    // later reused as t2_rb [18000,256]
  float* wtsb  = w; w += SZ_RB;
  float* msob  = w; w += SZ_RB;
  float* accb  = w; w += SZ_X;
  float* cntb  = w; w += 5008;      // contiguous after accb (single memset)
  float* finb  = w; w += SZ_X;

  const float* cur = tgt_in;
  for (int l = 0; l < 4; ++l) {
    const float* inw  = P_inw  + (size_t)l * 768 * 256;
    const float* inb  = P_inb  + (size_t)l * 768;
    const float* aow  = P_aow  + (size_t)l * 256 * 256;
    const float* aob  = P_aob  + (size_t)l * 256;
    const float* n2g  = P_n2g  + (size_t)l * 256;
    const float* n2b  = P_n2b  + (size_t)l * 256;
    const float* Wv   = P_Wv   + (size_t)l * 256 * 256;
    const float* bv   = P_bv   + (size_t)l * 256;
    const float* Woff = P_Woff + (size_t)l * 512 * 256;
    const float* boff = P_boff + (size_t)l * 512;
    const float* Ww   = P_Ww   + (size_t)l * 256 * 256;
    const float* bw   = P_bw   + (size_t)l * 256;
    const float* Wout = P_Wout + (size_t)l * 256 * 256;
    const float* bout = P_bout + (size_t)l * 256;
    const float* n1g  = P_n1g  + (size_t)l * 256;
    const float* n1b  = P_n1b  + (size_t)l * 256;
    const float* W1   = P_W1   + (size_t)l * 1024 * 256;
    const float* b1   = P_b1   + (size_t)l * 1024;
    const float* W2   = P_W2   + (size_t)l * 256 * 1024;
    const float* b2   = P_b2   + (size_t)l * 256;
    const float* n3g  = P_n3g  + (size_t)l * 256;
    const float* n3b  = P_n3b  + (size_t)l * 256;

    // ---- global self-attention ----
    add_pos<<<MROWS, 256, 0, stream>>>(cur, qpos, xb);
    // q,k from (tgt+pos); v from tgt — strided panels of the [.,768] buffer
    launch_gemm(xb,  inw,             inb,       qkvb,       MROWS, 512, 256, 768, 0, stream);
    launch_gemm(cur, inw + 512 * 256, inb + 512, qkvb + 512, MROWS, 256, 256, 768, 0, stream);
    mha_attn<<<dim3(CTILES, 16), 256, 0, stream>>>(qkvb, attno);
    launch_gemm(attno, aow, aob, t2s, MROWS, 256, 256, 256, 0, stream);
    residual_ln<<<(MROWS + 7) / 8, 256, 0, stream>>>(cur, t2s, n2g, n2b, xb, MROWS);

    // ---- per-camera rebatch + deformable cross-attention ----
    gather_rb<<<RBROWS, 256, 0, stream>>>(xb, qpos, refp, cidx, tgtrb, qrb, refrb);
    launch_gemm(src, Wv,   bv,   valb, BN_ * LIN_, 256, 256, 256, 0, stream);
    launch_gemm(qrb, Woff, boff, offb, RBROWS,     512, 256, 512, 0, stream);
    launch_gemm(qrb, Ww,   bw,   wtsb, RBROWS,     256, 256, 256, 0, stream);
    deform_sample<<<RBROWS, 256, 0, stream>>>(valb, offb, wtsb, refrb, msob);
    launch_gemm(msob, Wout, bout, offb /*t2_rb*/, RBROWS, 256, 256, 256, 0, stream);
    residual_ln<<<(RBROWS + 7) / 8, 256, 0, stream>>>(tgtrb, offb, n1g, n1b,
                                                      qrb /*rb2*/, RBROWS);

    // ---- scatter back, average across cameras ----
    hipMemsetAsync(accb, 0, (SZ_X + 5008) * sizeof(float), stream);
    scatter_add<<<RBROWS, 256, 0, stream>>>(qrb, cidx, accb, cntb);
    normalize_avg<<<MROWS, 256, 0, stream>>>(accb, cntb, finb);

    // ---- FFN ----
    launch_gemm(finb, W1, b1, valb /*ffn hidden*/, MROWS, 1024, 256, 1024, 1, stream);
    launch_gemm(valb, W2, b2, t2s, MROWS, 256, 1024, 256, 0, stream);
    float* outT = (l == 3) ? (float*)d_out : ((l & 1) ? tgtB : tgtA);
    residual_ln<<<(MROWS + 7) / 8, 256, 0, stream>>>(finb, t2s, n3g, n3b, outT, MROWS);
    cur = outT;
  }
}